// MyronenkoConvolutionBlock_32066225832300
// MI455X (gfx1250) — compile-verified
//
#include <hip/hip_runtime.h>
#include <stdint.h>

typedef __attribute__((ext_vector_type(2))) float        v2f;
typedef __attribute__((ext_vector_type(8))) float        v8f;
typedef __attribute__((ext_vector_type(4))) unsigned int v4u;
typedef __attribute__((ext_vector_type(4))) int          v4i;
typedef __attribute__((ext_vector_type(8))) int          v8i;

#define ELEMS_TOTAL 33554432u   // 2*16*64*128*128
#define GROUP_ELEMS 2097152u    // 2*64*128*128 (one (b,group) chunk, contiguous)
#define HW          16384u      // 128*128
#define DHW         1048576u    // 64*128*128
#define KTOT        432u        // 16 ic * 27 taps
#define NWGT        6912u       // 16 oc * 432
#define NBINS       21

// padded activation tensor, channel-PAIR interleaved:
//   [b(2)][pair(8)][66][130][130][2]   (pair p holds channels 2p, 2p+1)
#define PW    130u
#define PHW   16900u            // 130*130
#define CS    1115400u          // 66*130*130 (per-channel elems)
#define PRS   2230800u          // 2*CS   (per channel-pair stride, elems)
#define PAD_TOTAL 35692800u     // 2*8*PRS = 32*CS

// ---------------------------------------------------------------------------
// Kernel 1: per-(b,group) partial sums (deterministic two-pass reduction)
// ---------------------------------------------------------------------------
__global__ void gn_reduce1(const float* __restrict__ x, float* __restrict__ part) {
    __shared__ float ss[256];
    __shared__ float sq[256];
    const unsigned bg  = blockIdx.x >> 7;
    const unsigned sub = blockIdx.x & 127u;
    const float* p = x + (size_t)bg * GROUP_ELEMS + (size_t)sub * 16384u;
    float s = 0.f, q = 0.f;
    for (unsigned i = threadIdx.x; i < 16384u; i += 256u) {
        float v = p[i];
        s += v;
        q += v * v;
    }
    ss[threadIdx.x] = s; sq[threadIdx.x] = q;
    __syncthreads();
    for (unsigned o = 128u; o > 0u; o >>= 1) {
        if (threadIdx.x < o) {
            ss[threadIdx.x] += ss[threadIdx.x + o];
            sq[threadIdx.x] += sq[threadIdx.x + o];
        }
        __syncthreads();
    }
    if (threadIdx.x == 0) {
        part[blockIdx.x * 2 + 0] = ss[0];
        part[blockIdx.x * 2 + 1] = sq[0];
    }
}

// ---------------------------------------------------------------------------
// Kernel 2: finalize mean / rstd per (b,group)
// ---------------------------------------------------------------------------
__global__ void gn_reduce2(const float* __restrict__ part, float* __restrict__ stats) {
    __shared__ float ss[128];
    __shared__ float sq[128];
    const unsigned bg = blockIdx.x;
    ss[threadIdx.x] = part[(bg * 128u + threadIdx.x) * 2 + 0];
    sq[threadIdx.x] = part[(bg * 128u + threadIdx.x) * 2 + 1];
    __syncthreads();
    for (unsigned o = 64u; o > 0u; o >>= 1) {
        if (threadIdx.x < o) {
            ss[threadIdx.x] += ss[threadIdx.x + o];
            sq[threadIdx.x] += sq[threadIdx.x + o];
        }
        __syncthreads();
    }
    if (threadIdx.x == 0) {
        const float inv  = 1.0f / (float)GROUP_ELEMS;
        const float mean = ss[0] * inv;
        const float var  = sq[0] * inv - mean * mean;
        stats[bg * 2 + 0] = mean;
        stats[bg * 2 + 1] = rsqrtf(var + 1e-5f);
    }
}

// ---------------------------------------------------------------------------
// Kernel 3a: zero the integer histogram bins
// ---------------------------------------------------------------------------
__global__ void hist_zero(unsigned int* __restrict__ bins) {
    if (threadIdx.x < NBINS) bins[threadIdx.x] = 0u;
}

// ---------------------------------------------------------------------------
// Kernel 3b: zero the padded activation buffer (provides conv zero-halo)
// ---------------------------------------------------------------------------
__global__ void pad_zero(float* __restrict__ p) {
    const unsigned i = blockIdx.x * 256u + threadIdx.x;
    if (i < PAD_TOTAL) p[i] = 0.0f;
}

// ---------------------------------------------------------------------------
// Kernel 4: fused normalize + ReLU + histogram + depth-threshold.
// Writes thresholded activations into the padded, pair-interleaved buffer.
// ---------------------------------------------------------------------------
__global__ void norm_thresh_hist(const float* __restrict__ x,
                                 const float* __restrict__ gw,
                                 const float* __restrict__ gb,
                                 const float* __restrict__ thr_in,
                                 const float* __restrict__ stats,
                                 float* __restrict__ xpad,
                                 unsigned int* __restrict__ bins) {
    __shared__ unsigned int lb[NBINS];
    if (threadIdx.x < NBINS) lb[threadIdx.x] = 0u;
    __syncthreads();

    const unsigned e = blockIdx.x * 256u + threadIdx.x;
    const unsigned w = e & 127u;
    const unsigned h = (e >> 7) & 127u;
    const unsigned d = (e >> 14) & 63u;
    const unsigned c = (e >> 20) & 15u;
    const unsigned b = e >> 24;
    const unsigned bg = b * 8u + (c >> 1);

    const float mean = stats[bg * 2 + 0];
    const float rstd = stats[bg * 2 + 1];
    const float sw   = gw[c];
    const float sb   = gb[c];

    const float xv = x[e];
    float y = (xv - mean) * rstd * sw + sb;
    y = fmaxf(y, 0.0f);

    // histogram of y*(SCALE*SCALE_FACTOR) = y*0.5
    const float ys = y * 0.5f;
    const float v  = ceilf(ys * (ys >= 0.f ? 1.f : 0.f)) +
                     floorf(ys * (ys <= 0.f ? 1.f : 0.f));
    const int idx = (int)v + 10;
    if (idx >= 0 && idx <= 2 * 10) atomicAdd(&lb[idx], 1u);

    // threshold along depth: out[0]=y[0]; out[d]=y[d-1]+f(y[d]-y[d-1])
    const float thr = thr_in[0];
    float outv;
    if (d == 0u) {
        outv = y;
    } else {
        const float xp = x[e - HW];
        float yp = (xp - mean) * rstd * sw + sb;
        yp = fmaxf(yp, 0.0f);
        float dd = (y - yp) / thr;
        dd = (fabsf(dd) > 1.0f) ? dd : 0.0f;
        outv = yp + dd * thr;
    }
    // scatter into padded interior, pair-interleaved: coords = orig + 1
    xpad[(size_t)(b * 8u + (c >> 1)) * PRS +
         ((size_t)(d + 1u) * PHW + (size_t)(h + 1u) * PW + (w + 1u)) * 2u +
         (c & 1u)] = outv;

    __syncthreads();
    if (threadIdx.x < NBINS) atomicAdd(&bins[threadIdx.x], lb[threadIdx.x]);
}

// ---------------------------------------------------------------------------
// Kernel 6: convert histogram counts -> percentage floats
// ---------------------------------------------------------------------------
__global__ void hist_final(const unsigned int* __restrict__ bins, float* __restrict__ o) {
    if (threadIdx.x < NBINS)
        o[threadIdx.x] = (float)bins[threadIdx.x] * (100.0f / (float)ELEMS_TOTAL);
}

// ---------------------------------------------------------------------------
// Kernel 5: 3x3x3 conv as implicit GEMM with V_WMMA_F32_16X16X4_F32.
// Per wave: 16(OC) x 16(W) output tile, 108 WMMAs fully unrolled.
// B fragments are contiguous float2 pairs (pair-interleaved layout) ->
// direct register-pair operands, literal-offset loads, no predication.
// Weights staged in LDS via Tensor Data Mover, transposed to [oc][tap][ic].
// ---------------------------------------------------------------------------
__global__ void __launch_bounds__(256)
conv3d_wmma(const float* __restrict__ xpad, const float* __restrict__ wgt,
            float* __restrict__ out) {
    __shared__ float lds_stage[NWGT];  // raw [oc][ic][tap]
    __shared__ float lds_w[NWGT];      // transposed [oc][tap][ic]

    // --- stage weights into LDS with the Tensor Data Mover (wave 0 issues) ---
    if (threadIdx.x < 32u) {
        const unsigned lds_off = (unsigned)(uintptr_t)(void*)&lds_stage[0];
        const unsigned long long ga = (unsigned long long)(uintptr_t)wgt;
        v4u g0;
        g0.x = 1u;                                                  // count=1
        g0.y = lds_off;                                             // lds_addr
        g0.z = (unsigned)(ga & 0xFFFFFFFFull);                      // global_addr lo
        g0.w = (unsigned)((ga >> 32) & 0x01FFFFFFull) | (2u << 30); // addr hi | type=2
        v8i g1;
        g1[0] = (int)(2u << 16);                                    // data_size=4B
        g1[1] = (int)((NWGT & 0xFFFFu) << 16);                      // tensor_dim0 lo16
        g1[2] = (int)(((NWGT >> 16) & 0xFFFFu) | (1u << 16));       // dim0 hi | dim1=1
        g1[3] = (int)(NWGT << 16);                                  // tile_dim0
        g1[4] = 1;                                                  // tile_dim1=1
        g1[5] = (int)NWGT;                                          // dim0_stride
        g1[6] = 0;
        g1[7] = 0;
        v4i z = (v4i)0;
#if __clang_major__ >= 23
        __builtin_amdgcn_tensor_load_to_lds(g0, g1, z, z, (v8i)0, 0);
#else
        __builtin_amdgcn_tensor_load_to_lds(g0, g1, z, z, 0);
#endif
        __builtin_amdgcn_s_wait_tensorcnt(0);
    }
    __syncthreads();

    // LDS->LDS transpose: [oc][ic*27+tap] -> [oc][tap*16+ic]
    for (unsigned i = threadIdx.x; i < NWGT; i += 256u) {
        const unsigned oc = i / 432u;
        const unsigned r  = i - oc * 432u;
        const unsigned ic = r / 27u;
        const unsigned tp = r - ic * 27u;
        lds_w[oc * 432u + tp * 16u + ic] = lds_stage[i];
    }
    __syncthreads();

    const unsigned lane = threadIdx.x & 31u;
    const unsigned wave = threadIdx.x >> 5;
    const unsigned t    = blockIdx.x * 8u + wave;
    const unsigned wt = t & 7u;
    const unsigned h  = (t >> 3) & 127u;
    const unsigned d  = (t >> 10) & 63u;
    const unsigned b  = t >> 16;

    const unsigned m  = lane & 15u;        // A row (oc) / B col (w)
    const unsigned hf = lane >> 4;         // lane half
    const unsigned kb = hf << 1;           // per-lane-half K offset {0,2}
    const unsigned w0 = wt << 4;

    // spatial base (padded coords, kd=kh=kw=0 tap), in interleaved elems
    const size_t soff = ((size_t)d * PHW + (size_t)h * PW + (w0 + m)) * 2u;

    v8f acc = {};
#pragma unroll
    for (unsigned cc = 0; cc < 4u; ++cc) {
        // channel pair this lane consumes: pp = cc*2 + half; ic = 2*pp + {0,1}
        const unsigned pp = cc * 2u + hf;
        const float* pb = xpad + (size_t)(b * 8u + pp) * PRS + soff;
        const float* aw = &lds_w[m * 432u + cc * 4u + kb];
#pragma unroll
        for (unsigned kd = 0; kd < 3u; ++kd) {
#pragma unroll
            for (unsigned kh = 0; kh < 3u; ++kh) {
#pragma unroll
                for (unsigned kw = 0; kw < 3u; ++kw) {
                    const unsigned tp  = (kd * 3u + kh) * 3u + kw;
                    const unsigned dt2 = (kd * PHW + kh * PW + kw) * 2u; // literal
                    v2f a  = *(const v2f*)&aw[tp * 16u];   // ds_load_b64
                    v2f bf = *(const v2f*)&pb[dt2];        // contiguous pair
                    acc = __builtin_amdgcn_wmma_f32_16x16x4_f32(
                        false, a, false, bf, (short)0, acc, false, false);
                }
            }
        }
    }

    // D layout: VGPR j -> oc = j + 8*half, col = lane&15
    float* ob = out + (size_t)b * 16u * DHW + (size_t)d * HW +
                (size_t)h * 128u + w0 + m;
    const unsigned mb = hf << 3;
#pragma unroll
    for (int j = 0; j < 8; ++j)
        ob[(size_t)(mb + (unsigned)j) * DHW] = acc[j];
}

// ---------------------------------------------------------------------------
extern "C" void kernel_launch(void* const* d_in, const int* in_sizes, int n_in,
                              void* d_out, int out_size, void* d_ws, size_t ws_size,
                              hipStream_t stream) {
    const float* x   = (const float*)d_in[0];
    const float* gw  = (const float*)d_in[1];
    const float* gb  = (const float*)d_in[2];
    const float* thr = (const float*)d_in[3];
    const float* cw  = (const float*)d_in[4];
    float* out = (float*)d_out;

    // workspace layout
    float* xpad        = (float*)d_ws;              // 35,692,800 floats (padded)
    float* part        = xpad + PAD_TOTAL;          // 4096 floats
    float* stats       = part + 4096;               // 32 floats
    unsigned int* bins = (unsigned int*)(stats + 32);

    gn_reduce1<<<16 * 128, 256, 0, stream>>>(x, part);
    gn_reduce2<<<16, 128, 0, stream>>>(part, stats);
    hist_zero<<<1, 32, 0, stream>>>(bins);
    pad_zero<<<(PAD_TOTAL + 255u) / 256u, 256, 0, stream>>>(xpad);
    norm_thresh_hist<<<ELEMS_TOTAL / 256u, 256, 0, stream>>>(x, gw, gb, thr, stats, xpad, bins);
    conv3d_wmma<<<131072 / 8, 256, 0, stream>>>(xpad, cw, out);
    hist_final<<<1, 32, 0, stream>>>(bins, out + ELEMS_TOTAL);
}